// MoE_v1_644245094609
// MI455X (gfx1250) — compile-verified
//
#include <hip/hip_runtime.h>

// CDNA5 / gfx1250: wave32, WMMA f32 16x16x4.
typedef float v2f __attribute__((ext_vector_type(2)));
typedef float v8f __attribute__((ext_vector_type(8)));

#define NEXPERT 128
#define HW      65536   // 256*256 pixels
#define PSTRIDE 28      // floats per expert in params (7*K, K=4)

// One wave handles a 16-gaussian (4 experts) x 64-pixel strip:
//   A (16x4 f32) = polynomial coefficients of the 16 gaussians
//   B (4x16 f32) = pixel feature columns [x^2, xy, y^2, x | y, 1, 0, 0]
//   D = A*B (+C)  -> 16x16 exponents, exp2 applied per lane, k-reduction
//   is within-lane over D rows (each lane owns one pixel column).
__global__ __launch_bounds__(256) void gmix_wmma_kernel(
    const float* __restrict__ params, float* __restrict__ out) {

    const int lane  = threadIdx.x & 31;
    const int wave  = threadIdx.x >> 5;
    const int m     = lane & 15;   // A: gaussian row | B/D: pixel column
    const int khalf = lane >> 4;   // which K pair this lane supplies

    const int eg0     = blockIdx.y << 2;                  // 4 experts per group
    const int pixbase = (blockIdx.x << 9) + (wave << 6);  // 512 px/block, 64/wave

    // ---------- Build A: coefficients for gaussian row m ----------
    // params layout per expert: mu_x[4], mu_y[4], w[4], Sig[4][2][2]
    const int eidx = eg0 + (m >> 2);
    const int k    = m & 3;
    const float* pp = params + eidx * PSTRIDE;
    const float mux = pp[k];
    const float muy = pp[4 + k];
    const float s00 = pp[12 + 4 * k];
    const float s10 = pp[14 + 4 * k];   // tril: s01 == 0
    const float s11 = pp[15 + 4 * k];

    // |Sig d|^2 = a*dx^2 + b*dx*dy + c*dy^2
    const float a = s00 * s00 + s10 * s10;
    const float b = 2.0f * s10 * s11;
    const float c = s11 * s11;
    const float S = -0.72134752044448170368f;   // -0.5 * log2(e), folded into coeffs
    const float v0 = S * a;
    const float v1 = S * b;
    const float v2 = S * c;
    const float v3 = -S * (2.0f * a * mux + b * muy);
    const float v4 = -S * (b * mux + 2.0f * c * muy);
    const float v5 = S * (a * mux * mux + b * mux * muy + c * muy * muy);

    // A 16x4 f32 layout (ISA 7.12.2): lanes 0-15 -> K=0,1 ; lanes 16-31 -> K=2,3
    v2f A1, A2;                       // WMMA #1: features 0..3, WMMA #2: features 4..7
    A1.x = khalf ? v2 : v0;
    A1.y = khalf ? v3 : v1;
    A2.x = khalf ? 0.0f : v4;
    A2.y = khalf ? 0.0f : v5;

    // ---------- Weights for the two experts this half-wave reduces ----------
    const int eA = eg0 + (khalf << 1);          // rows 0-3 (+8*khalf)
    const float* pwA = params + eA * PSTRIDE + 8;
    const float wA0 = pwA[0], wA1 = pwA[1], wA2 = pwA[2], wA3 = pwA[3];
    const float* pwB = pwA + PSTRIDE;           // expert eA+1, rows 4-7 (+8*khalf)
    const float wB0 = pwB[0], wB1 = pwB[1], wB2 = pwB[2], wB3 = pwB[3];

    const float inv255 = 1.0f / 255.0f;

#pragma unroll
    for (int t = 0; t < 4; ++t) {
        // flat pixel p: grid[p] = (x = (p>>8)/255, y = (p&255)/255)
        const int p = pixbase + (t << 4) + m;
        const float x = (float)(p >> 8) * inv255;
        const float y = (float)(p & 255) * inv255;

        // B 4x16 f32: lane = pixel column; lanes 0-15 rows K=0,1; 16-31 rows K=2,3
        v2f B1, B2;
        B1.x = khalf ? y * y : x * x;
        B1.y = khalf ? x     : x * y;
        B2.x = khalf ? 0.0f  : y;
        B2.y = khalf ? 0.0f  : 1.0f;

        // exponent tile: D[g][pix] = -0.5*log2(e) * q(g, pix)
        v8f acc = {};
        acc = __builtin_amdgcn_wmma_f32_16x16x4_f32(
            false, A2, false, B2, (short)0, acc, false, false);
        acc = __builtin_amdgcn_wmma_f32_16x16x4_f32(
            false, A1, false, B1, (short)0, acc, false, false);

        // e = exp2(acc) ; rows 0-3 -> expert eA, rows 4-7 -> expert eA+1
        const float e0 = __builtin_amdgcn_exp2f(acc[0]);
        const float e1 = __builtin_amdgcn_exp2f(acc[1]);
        const float e2 = __builtin_amdgcn_exp2f(acc[2]);
        const float e3 = __builtin_amdgcn_exp2f(acc[3]);
        const float e4 = __builtin_amdgcn_exp2f(acc[4]);
        const float e5 = __builtin_amdgcn_exp2f(acc[5]);
        const float e6 = __builtin_amdgcn_exp2f(acc[6]);
        const float e7 = __builtin_amdgcn_exp2f(acc[7]);

        const float gA = (e0 + e1) + (e2 + e3);
        const float nA = e0 * wA0 + e1 * wA1 + e2 * wA2 + e3 * wA3;
        float yA = nA * __builtin_amdgcn_rcpf(fmaxf(gA, 1e-7f));
        yA = fminf(fmaxf(yA, 0.0f), 1.0f);

        const float gB = (e4 + e5) + (e6 + e7);
        const float nB = e4 * wB0 + e5 * wB1 + e6 * wB2 + e7 * wB3;
        float yB = nB * __builtin_amdgcn_rcpf(fmaxf(gB, 1e-7f));
        yB = fminf(fmaxf(yB, 0.0f), 1.0f);

        // lanes within a half-wave write 16 consecutive floats per expert
        out[eA * HW + p]       = yA;
        out[(eA + 1) * HW + p] = yB;
    }
}

extern "C" void kernel_launch(void* const* d_in, const int* in_sizes, int n_in,
                              void* d_out, int out_size, void* d_ws, size_t ws_size,
                              hipStream_t stream) {
    (void)in_sizes; (void)n_in; (void)d_ws; (void)ws_size; (void)out_size;
    // inputs: d_in[0]=height(=256), d_in[1]=width(=256), d_in[2]=params (128*1*28 f32)
    const float* params = (const float*)d_in[2];
    float* out = (float*)d_out;

    dim3 grid(128, 32, 1);   // 128 pixel-blocks (512 px each) x 32 expert groups
    gmix_wmma_kernel<<<grid, dim3(256, 1, 1), 0, stream>>>(params, out);
}